// GNNEncoder_88132728914199
// MI455X (gfx1250) — compile-verified
//
#include <hip/hip_runtime.h>
#include <math.h>

typedef __attribute__((ext_vector_type(2))) float v2f;
typedef __attribute__((ext_vector_type(8))) float v8f;

// ==================================================================
// CSR build (graph is shared by all 3 layers; built once per launch)
// ==================================================================
__global__ void deg_init(int* __restrict__ deg, int N) {
    int i = blockIdx.x * blockDim.x + threadIdx.x;
    if (i < N) deg[i] = 1;                       // self-loop
}

__global__ void deg_count(const int* __restrict__ dsts, long long E, int* __restrict__ deg) {
    long long e = (long long)blockIdx.x * blockDim.x + threadIdx.x;
    if (e < E) atomicAdd(&deg[dsts[e]], 1);
}

// block-wise inclusive scan (256 per block) + per-block totals
__global__ void scan1(const int* __restrict__ deg, int* __restrict__ inc,
                      int* __restrict__ bsum, int N) {
    __shared__ int sh[256];
    int i = blockIdx.x * 256 + threadIdx.x;
    sh[threadIdx.x] = (i < N) ? deg[i] : 0;
    __syncthreads();
    for (int off = 1; off < 256; off <<= 1) {
        int t = (threadIdx.x >= off) ? sh[threadIdx.x - off] : 0;
        __syncthreads();
        sh[threadIdx.x] += t;
        __syncthreads();
    }
    if (i < N) inc[i] = sh[threadIdx.x];
    if (threadIdx.x == 255) bsum[blockIdx.x] = sh[255];
}

// single-block inclusive scan of the block sums (nb <= 1024)
__global__ void scan2(int* __restrict__ bsum, int nb) {
    __shared__ int sh[1024];
    sh[threadIdx.x] = (threadIdx.x < nb) ? bsum[threadIdx.x] : 0;
    __syncthreads();
    for (int off = 1; off < 1024; off <<= 1) {
        int t = (threadIdx.x >= off) ? sh[threadIdx.x - off] : 0;
        __syncthreads();
        sh[threadIdx.x] += t;
        __syncthreads();
    }
    if (threadIdx.x < nb) bsum[threadIdx.x] = sh[threadIdx.x];
}

// rowptr[i+1] = inclusive_scan(deg)[i];  rowptr[0] = 0
__global__ void scan3(const int* __restrict__ inc, const int* __restrict__ bsum,
                      int* __restrict__ rowptr, int N) {
    int i = blockIdx.x * 256 + threadIdx.x;
    if (i < N) {
        int add = (blockIdx.x > 0) ? bsum[blockIdx.x - 1] : 0;
        rowptr[i + 1] = inc[i] + add;
    }
    if (i == 0) rowptr[0] = 0;
}

__global__ void cursor_init(int* __restrict__ cur, const int* __restrict__ rowptr, int N) {
    int i = blockIdx.x * blockDim.x + threadIdx.x;
    if (i < N) cur[i] = rowptr[i];
}

// scatter src ids into dst-sorted adjacency (self-loops appended)
__global__ void csr_fill(const int* __restrict__ srcs, const int* __restrict__ dsts,
                         long long E, int N, int* __restrict__ cur, int* __restrict__ col) {
    long long e = (long long)blockIdx.x * blockDim.x + threadIdx.x;
    if (e < E) {
        int pos = atomicAdd(&cur[dsts[e]], 1);
        col[pos] = srcs[e];
    } else if (e < E + N) {
        int n = (int)(e - E);
        int pos = atomicAdd(&cur[n], 1);
        col[pos] = n;                            // self-loop
    }
}

// ==================================================================
// Dense transform: h = in[N,FIN] @ W[FIN,64] via f32 WMMA 16x16x4,
// plus attention logits asrc/adst. One wave per 16-row tile.
//   A: lane L, vgpr v -> A[L%16][k0 + 2*(L/16) + v]
//   B: lane L, vgpr v -> B[k0 + 2*(L/16) + v][L%16]
//   D: lane L, vgpr v -> D[v + 8*(L/16)][L%16]
// ==================================================================
template<int FIN, int H>
__global__ __launch_bounds__(32) void gat_transform(
    const float* __restrict__ in, const float* __restrict__ W,
    const float* __restrict__ a_src, const float* __restrict__ a_dst,
    float* __restrict__ hout, float* __restrict__ asrc,
    float* __restrict__ adst, int N)
{
    __shared__ float lds[16 * 64];
    const int lane = threadIdx.x;
    const int half = lane >> 4;
    const int l16  = lane & 15;
    const int row0 = blockIdx.x * 16;
    int ar = row0 + l16;  if (ar >= N) ar = N - 1;     // clamp, no divergence

    v8f acc[4] = {};
    for (int k0 = 0; k0 < FIN; k0 += 4) {
        const float* ap = in + (size_t)ar * FIN + k0 + 2 * half;
        v2f A;  A.x = ap[0];  A.y = ap[1];
#pragma unroll
        for (int c = 0; c < 4; ++c) {
            const float* bp = W + (size_t)(k0 + 2 * half) * 64 + c * 16 + l16;
            v2f B;  B.x = bp[0];  B.y = bp[64];
            acc[c] = __builtin_amdgcn_wmma_f32_16x16x4_f32(
                false, A, false, B, (short)0, acc[c], false, false);
        }
    }

#pragma unroll
    for (int c = 0; c < 4; ++c)
#pragma unroll
        for (int v = 0; v < 8; ++v)
            lds[(v + 8 * half) * 64 + c * 16 + l16] = acc[c][v];
    __syncthreads();

    for (int j = 0; j < 32; ++j) {                      // coalesced tile store
        int idx = j * 32 + lane;
        int r = idx >> 6, cc = idx & 63;
        if (row0 + r < N)
            hout[(size_t)(row0 + r) * 64 + cc] = lds[idx];
    }

    if (H == 8) {
        for (int i = 0; i < 4; ++i) {
            int p = lane + 32 * i;                      // 16 rows x 8 heads
            int r = p >> 3, hh = p & 7;
            float s1 = 0.0f, s2 = 0.0f;
            for (int d = 0; d < 8; ++d) {
                float hv = lds[r * 64 + hh * 8 + d];
                s1 += hv * a_src[hh * 8 + d];
                s2 += hv * a_dst[hh * 8 + d];
            }
            if (row0 + r < N) {
                asrc[(size_t)(row0 + r) * 8 + hh] = s1;
                adst[(size_t)(row0 + r) * 8 + hh] = s2;
            }
        }
    } else {  // H == 1, D = 64
        if (lane < 16 && row0 + lane < N) {
            float s1 = 0.0f, s2 = 0.0f;
            for (int d = 0; d < 64; ++d) {
                float hv = lds[lane * 64 + d];
                s1 += hv * a_src[d];
                s2 += hv * a_dst[d];
            }
            asrc[row0 + lane] = s1;
            adst[row0 + lane] = s2;
        }
    }
}

// ==================================================================
// Atomic-free gather aggregation: one wave per destination node.
// Lane owns channels {lane, lane+32}; accumulates numerator AND its
// head's softmax denominator in registers. Head weights computed once
// per neighbor and broadcast via wave32 shuffle. Softmax max-shift
// dropped (shift-invariant; |e| << 88 here). Bias/ReLU fused.
// ==================================================================
template<int H, bool RELU>
__global__ __launch_bounds__(256) void gat_gather(
    const int* __restrict__ rowptr, const int* __restrict__ col,
    const float* __restrict__ hbuf, const float* __restrict__ asrc,
    const float* __restrict__ adst, const float* __restrict__ bias,
    float* __restrict__ out, int N)
{
    int n = blockIdx.x * (blockDim.x >> 5) + (threadIdx.x >> 5);
    if (n >= N) return;                                 // uniform per wave
    const int lane = threadIdx.x & 31;
    const int c0 = lane, c1 = lane + 32;

    float adme = (H == 8) ? adst[(size_t)n * 8 + (lane & 7)] : adst[n];
    const int beg = rowptr[n], end = rowptr[n + 1];

    float a0 = 0.0f, a1 = 0.0f, d0 = 0.0f, d1 = 0.0f;
    for (int j = beg; j < end; ++j) {
        int s = col[j];
        float w0, w1;
        if (H == 8) {
            float ev = asrc[(size_t)s * 8 + (lane & 7)] + adme;
            ev = (ev > 0.0f) ? ev : 0.2f * ev;          // leaky_relu(0.2)
            float wl = expf(ev);                        // lane g holds w[g&7]
            w0 = __shfl(wl, lane >> 3, 32);             // head of c0
            w1 = __shfl(wl, (lane >> 3) + 4, 32);       // head of c1
        } else {
            float ev = asrc[s] + adme;
            ev = (ev > 0.0f) ? ev : 0.2f * ev;
            w0 = expf(ev);  w1 = w0;
        }
        const float* hr = hbuf + (size_t)s * 64;
        a0 += w0 * hr[c0];  d0 += w0;
        a1 += w1 * hr[c1];  d1 += w1;
    }

    float o0 = a0 / d0 + bias[c0];
    float o1 = a1 / d1 + bias[c1];
    if (RELU) { o0 = fmaxf(o0, 0.0f);  o1 = fmaxf(o1, 0.0f); }
    out[(size_t)n * 64 + c0] = o0;
    out[(size_t)n * 64 + c1] = o1;
}

// batch passthrough (int -> float, per harness output convention)
__global__ void batch_copy(const int* __restrict__ b, float* __restrict__ out, int N) {
    int i = blockIdx.x * blockDim.x + threadIdx.x;
    if (i < N) out[i] = (float)b[i];
}

// ==================================================================
extern "C" void kernel_launch(void* const* d_in, const int* in_sizes, int n_in,
                              void* d_out, int out_size, void* d_ws, size_t ws_size,
                              hipStream_t stream)
{
    const float* x    = (const float*)d_in[0];
    const int*   edge = (const int*)d_in[1];     // [2,E]: row0=src, row1=dst
    const int*   bat  = (const int*)d_in[2];
    const float* W1   = (const float*)d_in[3];
    const float* a1s  = (const float*)d_in[4];
    const float* a1d  = (const float*)d_in[5];
    const float* b1   = (const float*)d_in[6];
    const float* W2   = (const float*)d_in[7];
    const float* a2s  = (const float*)d_in[8];
    const float* a2d  = (const float*)d_in[9];
    const float* b2   = (const float*)d_in[10];
    const float* W3   = (const float*)d_in[11];
    const float* a3s  = (const float*)d_in[12];
    const float* a3d  = (const float*)d_in[13];
    const float* b3   = (const float*)d_in[14];

    const int N = in_sizes[0] / 8;
    const long long E = in_sizes[1] / 2;
    const int* srcs = edge;
    const int* dsts = edge + E;

    // ---- workspace layout ----
    float* P    = (float*)d_ws;                  // ping buffer  [N*64]
    float* Hb   = P    + (size_t)N * 64;         // h            [N*64]
    float* ASRC = Hb   + (size_t)N * 64;         // [N*8]
    float* ADST = ASRC + (size_t)N * 8;          // [N*8]
    int*   rowptr = (int*)(ADST + (size_t)N * 8);     // [N+1]
    int*   deg    = rowptr + (N + 1);                 // [N] (reused as cursor)
    int*   inc    = deg + N;                          // [N]
    int*   bsum   = inc + N;                          // [nb]
    const int nb  = (N + 255) / 256;
    int*   col    = bsum + ((nb + 63) & ~63);         // [E+N]
    (void)ws_size;

    const long long Etot = E + N;
    const int eblocks = (int)((Etot + 255) / 256);
    const int nblocks = (N + 255) / 256;
    const int tiles   = (N + 15) / 16;
    const int gblocks = (N + 7) / 8;             // 8 waves (nodes) per block

    // ---- build dst-sorted CSR once (reused by all 3 layers) ----
    deg_init   <<<nblocks, 256, 0, stream>>>(deg, N);
    deg_count  <<<(int)((E + 255) / 256), 256, 0, stream>>>(dsts, E, deg);
    scan1      <<<nb, 256, 0, stream>>>(deg, inc, bsum, N);
    scan2      <<<1, 1024, 0, stream>>>(bsum, nb);
    scan3      <<<nb, 256, 0, stream>>>(inc, bsum, rowptr, N);
    cursor_init<<<nblocks, 256, 0, stream>>>(deg, rowptr, N);
    csr_fill   <<<eblocks, 256, 0, stream>>>(srcs, dsts, E, N, deg, col);

    float* out_h = (float*)d_out;

    // ---- Layer 1: FIN=8, H=8, ReLU ----
    gat_transform<8, 8><<<tiles, 32, 0, stream>>>(x, W1, a1s, a1d, Hb, ASRC, ADST, N);
    gat_gather<8, true><<<gblocks, 256, 0, stream>>>(rowptr, col, Hb, ASRC, ADST, b1, P, N);

    // ---- Layer 2: FIN=64, H=8, ReLU ----
    gat_transform<64, 8><<<tiles, 32, 0, stream>>>(P, W2, a2s, a2d, Hb, ASRC, ADST, N);
    gat_gather<8, true><<<gblocks, 256, 0, stream>>>(rowptr, col, Hb, ASRC, ADST, b2, P, N);

    // ---- Layer 3: FIN=64, H=1, no ReLU, writes d_out ----
    gat_transform<64, 1><<<tiles, 32, 0, stream>>>(P, W3, a3s, a3d, Hb, ASRC, ADST, N);
    gat_gather<1, false><<<gblocks, 256, 0, stream>>>(rowptr, col, Hb, ASRC, ADST, b3, out_h, N);

    // ---- batch passthrough ----
    if (out_size >= N * 64 + N)
        batch_copy<<<(N + 255) / 256, 256, 0, stream>>>(bat, out_h + (size_t)N * 64, N);
}